// SymplecticGNNKalmanLayer_82497731821680
// MI455X (gfx1250) — compile-verified
//
#include <hip/hip_runtime.h>
#include <cstdint>

// ---------------------------------------------------------------------------
// SymplecticGNNKalmanLayer for MI455X (gfx1250).
// Memory/atomic-bound scatter-add: no matmul -> no WMMA. CDNA5 paths used:
//   - TENSOR_LOAD_TO_LDS (TDM DMA) stages each edge tile into LDS exactly once
//   - s_wait_tensorcnt synchronization (TENSORcnt)
//   - coalesced b128 gathers + fp32 global atomics (L2-resident working set)
// ---------------------------------------------------------------------------

#define DT_STEP 0.01f
#define TILE_E   256              // edges staged per block (2 KB of LDS)
#define SCATTER_BLOCK 256
#define DFEAT    64               // feature dim (fixed by reference)

typedef __attribute__((ext_vector_type(4))) unsigned int v4u;
typedef __attribute__((ext_vector_type(8))) int          v8i;
typedef __attribute__((ext_vector_type(4))) int          v4i;

typedef __attribute__((address_space(3))) int lds_int_t;

// ---------------------------------------------------------------------------
// Kernel 1: msg[b,dst,:] += q[b,dst,:] - q[b,src,:]   (scatter-add via atomics)
// grid = ceil(E/TILE_E) blocks, 256 threads (8 wave32). Each block stages one
// edge tile via TDM, then loops over all batches reusing the staged tile.
// ---------------------------------------------------------------------------
__global__ void __launch_bounds__(SCATTER_BLOCK)
scatter_kernel(const float* __restrict__ q,
               const int*   __restrict__ edges,
               float*       __restrict__ msg,
               int N, int E, int B)
{
    __shared__ int s_edges[TILE_E * 2];

    const int e0   = blockIdx.x * TILE_E;
    const int nloc = (E - e0 < TILE_E) ? (E - e0) : TILE_E;

    // ---- TDM: DMA this tile of edge pairs (nloc*2 ints) into LDS once -----
    if (threadIdx.x == 0) {
        const uint64_t ga      = (uint64_t)(uintptr_t)(edges + (size_t)e0 * 2);
        const uint32_t lds_off = (uint32_t)(uintptr_t)(lds_int_t*)s_edges;
        const uint32_t elems   = (uint32_t)(nloc * 2);    // 4-byte elements

        // D# group 0 (128b): count=1 | lds_addr | global_addr | type=2
        v4u g0;
        g0.x = 1u;                                        // count = 1 (valid)
        g0.y = lds_off;                                   // LDS byte address
        g0.z = (uint32_t)ga;                              // global_addr[31:0]
        g0.w = ((uint32_t)(ga >> 32) & 0x01FFFFFFu)       // global_addr[56:32]
             | (2u << 30);                                // type = 2 ("image")

        // D# group 1 (256b): 1-row tile of `elems` 4-byte elements
        v8i g1;
        g1[0] = (int)(2u << 16);                          // data_size = 4 bytes
        g1[1] = (int)((elems & 0xFFFFu) << 16);           // tensor_dim0[15:0]
        g1[2] = (int)((elems >> 16) | (1u << 16));        // tensor_dim0[31:16] | tensor_dim1=1
        g1[3] = (int)(elems << 16);                       // tile_dim0
        g1[4] = 1;                                        // tile_dim1 = 1
        g1[5] = (int)elems;                               // tensor_dim0_stride[31:0]
        g1[6] = 0;
        g1[7] = 0;

        v4i gz4 = {0, 0, 0, 0};                           // groups 2/3 unused (<=2D)
        v8i gz8 = {0, 0, 0, 0, 0, 0, 0, 0};               // extra group (clang-23 form)
        __builtin_amdgcn_tensor_load_to_lds(g0, g1, gz4, gz4, gz8, 0);
        __builtin_amdgcn_s_wait_tensorcnt(0);             // TENSORcnt == 0
    }
    __syncthreads();

    // ---- scatter: 16 lanes per edge, one float4 feature group per lane ----
    for (int b = 0; b < B; ++b) {
        const size_t baseQ = (size_t)b * (size_t)N * DFEAT;

        #pragma unroll
        for (int it = 0; it < (TILE_E * (DFEAT / 4)) / SCATTER_BLOCK; ++it) {
            const int task = threadIdx.x + it * SCATTER_BLOCK;
            const int el   = task >> 4;       // edge within tile
            const int g    = task & 15;       // float4 group within feature row
            if (el < nloc) {
                const int src = s_edges[el * 2 + 0];
                const int dst = s_edges[el * 2 + 1];
                const float4 a = ((const float4*)(q + baseQ + (size_t)src * DFEAT))[g];
                const float4 c = ((const float4*)(q + baseQ + (size_t)dst * DFEAT))[g];
                float* o = msg + baseQ + (size_t)dst * DFEAT + (size_t)g * 4;
                atomicAdd(o + 0, c.x - a.x);
                atomicAdd(o + 1, c.y - a.y);
                atomicAdd(o + 2, c.z - a.z);
                atomicAdd(o + 3, c.w - a.w);
            }
        }
    }
}

// ---------------------------------------------------------------------------
// Kernel 2: fused leapfrog update + node-0 Kalman correction (division-free).
//   grid = (ceil(N*16/256), B); j = in-batch float4 index; node = j>>4.
//   q_new = q + DT*p + 0.5*DT^2*msg
//   p_new = p + DT*msg        (msg lives in pout[] and is overwritten in place)
//   node 0: inn = obs - q_new; q_new += kq*inn; p_new += kp*inn
// ---------------------------------------------------------------------------
__global__ void __launch_bounds__(256)
finalize_kernel(const float* __restrict__ q,
                const float* __restrict__ p,
                const float* __restrict__ obs,
                const float* __restrict__ kg,
                float* __restrict__ qout,
                float* __restrict__ pout,
                int N)
{
    const int j = blockIdx.x * 256 + threadIdx.x;    // float4 index within batch
    const int groups = N * (DFEAT / 4);
    if (j >= groups) return;

    const int b = blockIdx.y;
    const size_t i = (size_t)b * (size_t)groups + (size_t)j;

    const float4 qv = ((const float4*)q)[i];
    const float4 pv = ((const float4*)p)[i];
    const float4 mv = ((const float4*)pout)[i];      // accumulated msg

    const float h = 0.5f * DT_STEP * DT_STEP;
    float4 qn, pn;
    qn.x = qv.x + DT_STEP * pv.x + h * mv.x;  pn.x = pv.x + DT_STEP * mv.x;
    qn.y = qv.y + DT_STEP * pv.y + h * mv.y;  pn.y = pv.y + DT_STEP * mv.y;
    qn.z = qv.z + DT_STEP * pv.z + h * mv.z;  pn.z = pv.z + DT_STEP * mv.z;
    qn.w = qv.w + DT_STEP * pv.w + h * mv.w;  pn.w = pv.w + DT_STEP * mv.w;

    if ((j >> 4) == 0) {                             // node 0 of this batch
        const int d = (j & 15) * 4;
        const float4 ov = *(const float4*)(obs + (size_t)b * DFEAT + d);
        const float4 kq = *(const float4*)(kg + d);
        const float4 kp = *(const float4*)(kg + DFEAT + d);
        float in0 = ov.x - qn.x; qn.x += kq.x * in0; pn.x += kp.x * in0;
        float in1 = ov.y - qn.y; qn.y += kq.y * in1; pn.y += kp.y * in1;
        float in2 = ov.z - qn.z; qn.z += kq.z * in2; pn.z += kp.z * in2;
        float in3 = ov.w - qn.w; qn.w += kq.w * in3; pn.w += kp.w * in3;
    }

    ((float4*)qout)[i] = qn;
    ((float4*)pout)[i] = pn;   // overwrites msg in place (thread-owned elements)
}

// ---------------------------------------------------------------------------
extern "C" void kernel_launch(void* const* d_in, const int* in_sizes, int n_in,
                              void* d_out, int out_size, void* d_ws, size_t ws_size,
                              hipStream_t stream)
{
    const float* q     = (const float*)d_in[0];   // (B,N,D)
    const float* p     = (const float*)d_in[1];   // (B,N,D)
    const float* obs   = (const float*)d_in[2];   // (B,D)
    const float* kg    = (const float*)d_in[3];   // (2D,)
    const int*   edges = (const int*)  d_in[4];   // (E,2)

    const int D = in_sizes[3] / 2;                       // 64
    const int B = in_sizes[2] / D;                       // 2
    const long long BND = (long long)in_sizes[0];        // B*N*D
    const int N = (int)(BND / ((long long)B * D));       // 20000
    const int E = in_sizes[4] / 2;                       // 320000

    float* qout = (float*)d_out;          // first half of output
    float* pout = qout + BND;             // second half; doubles as msg accumulator

    // zero the msg accumulator (graph-capture-legal memset node)
    (void)hipMemsetAsync(pout, 0, (size_t)BND * sizeof(float), stream);

    scatter_kernel<<<dim3((unsigned)((E + TILE_E - 1) / TILE_E)),
                     SCATTER_BLOCK, 0, stream>>>(q, edges, pout, N, E, B);

    const int groups = N * (D / 4);                      // float4 groups per batch
    dim3 fgrid((unsigned)((groups + 255) / 256), (unsigned)B);
    finalize_kernel<<<fgrid, 256, 0, stream>>>(q, p, obs, kg, qout, pout, N);
}